// WindowedMaskedAttentionCalculation_72189810311255
// MI455X (gfx1250) — compile-verified
//
#include <hip/hip_runtime.h>

typedef float v2f __attribute__((ext_vector_type(2)));
typedef float v8f __attribute__((ext_vector_type(8)));

// One workgroup = 4 waves (128 threads), handles 64 rows of I for one (b,p,h).
// Each wave computes a 16(i) x 256(j) score strip with v_wmma_f32_16x16x4_f32,
// does a full-row softmax in registers, then attn@V with the same WMMA op.
__global__ __launch_bounds__(128)
void wmattn_gfx1250_kernel(const float* __restrict__ q,
                           const float* __restrict__ k,
                           const float* __restrict__ v,
                           const float* __restrict__ pos,
                           const float* __restrict__ msk,
                           float* __restrict__ out)
{
    __shared__ float lds[32768];          // 128 KB (CDNA5 WGP LDS = 320 KB)
    float* ldsK = lds;                    // 64 KB: K tile, later reused for attn
    float* ldsV = lds + 16384;            // 64 KB: V tile

    const int tid  = threadIdx.x;
    const int wave = tid >> 5;            // wave32
    const int lane = tid & 31;
    const int lq   = lane & 15;           // row/col index inside a 16-frag
    const int half = lane >> 4;           // selects K-pair {0,1} vs {2,3}

    const int blk   = blockIdx.x;
    const int itile = blk & 3;            // 64-row slice of I
    const int bph   = blk >> 2;           // fused (b,p,h)

    const float* Q  = q   + (size_t)bph * (256 * 64);
    const float* Kg = k   + (size_t)bph * (256 * 64);
    const float* Vg = v   + (size_t)bph * (256 * 64);
    const float* Pg = pos + (size_t)bph * (256 * 256);
    const float* Mg = msk + (size_t)bph * (256 * 256);
    float*       Og = out + (size_t)bph * (256 * 64);

    // ---- stage K and V into LDS (b128 copies) ----
    {
        const float4* gk = (const float4*)Kg;
        const float4* gv = (const float4*)Vg;
        float4* sk = (float4*)ldsK;
        float4* sv = (float4*)ldsV;
        for (int c = tid; c < 4096; c += 128) {
            sk[c] = gk[c];
            sv[c] = gv[c];
        }
    }
    __syncthreads();

    const int i0 = itile * 64 + wave * 16;   // this wave's first I row

    // ---- Q fragments in 16x4 f32 A-layout, pre-scaled by 1/sqrt(D) ----
    // lanes 0-15 = rows, half selects K {2h, 2h+1} inside each 4-wide K step.
    v2f a[16];
    {
        const float* qrow = Q + (size_t)(i0 + lq) * 64 + half * 2;
        #pragma unroll
        for (int kk = 0; kk < 16; ++kk) {
            v2f t = *(const v2f*)(qrow + kk * 4);
            a[kk] = t * 0.125f;              // INV_TEMPERATURE folded into Q
        }
    }

    // ---- S = mask * (Q K^T * invT + pos), accumulated in C-layout ----
    // Software pipeline: issue the 16 global bias/mask loads and the 8 batched
    // LDS fragment loads BEFORE the dependent 16-deep WMMA chain, so the chain
    // covers their latency; one wait per tile instead of one per element.
    v8f S[16];
    #pragma unroll
    for (int t = 0; t < 16; ++t) {
        // (1) bias/mask loads for THIS tile — independent of the wmma chain
        float pv[8], mv[8];
        #pragma unroll
        for (int r = 0; r < 8; ++r) {
            size_t off = (size_t)(i0 + r + half * 8) * 256 + t * 16 + lq;
            pv[r] = Pg[off];
            mv[r] = Mg[off];
        }
        // prefetch next tile's bias/mask stream into near caches
        {
            size_t poff = (size_t)(i0 + half * 8) * 256 + ((t + 1) & 15) * 16 + lq;
            __builtin_prefetch(Pg + poff, 0, 3);
            __builtin_prefetch(Mg + poff, 0, 3);
        }
        // (2) all 16 B-fragments for this tile in one batch (ds_load_2addr_b64)
        v2f bf[16];
        const float* kb = ldsK + (size_t)(t * 16 + lq) * 64 + half * 2;
        #pragma unroll
        for (int kk = 0; kk < 16; ++kk) {
            bf[kk] = *(const v2f*)(kb + kk * 4);
        }
        // (3) dependent WMMA chain
        v8f acc = {};
        #pragma unroll
        for (int kk = 0; kk < 16; ++kk) {
            acc = __builtin_amdgcn_wmma_f32_16x16x4_f32(
                false, a[kk], false, bf[kk], (short)0, acc, false, false);
        }
        // (4) combine with bias/mask (loads have had the whole chain to land)
        #pragma unroll
        for (int r = 0; r < 8; ++r) {
            acc[r] = mv[r] * (acc[r] + pv[r]);
        }
        S[t] = acc;
    }

    // ---- softmax over all 256 j, fully in registers ----
    // xor-shuffles 1/2/4/8 reduce within each 16-lane half (matches C striping)
    v8f rinv;
    #pragma unroll
    for (int r = 0; r < 8; ++r) {
        float mx = S[0][r];
        #pragma unroll
        for (int t = 1; t < 16; ++t) mx = fmaxf(mx, S[t][r]);
        #pragma unroll
        for (int d = 1; d <= 8; d <<= 1) mx = fmaxf(mx, __shfl_xor(mx, d, 32));
        float sm = 0.0f;
        #pragma unroll
        for (int t = 0; t < 16; ++t) {
            float e = __expf(S[t][r] - mx);
            S[t][r] = e;
            sm += e;
        }
        #pragma unroll
        for (int d = 1; d <= 8; d <<= 1) sm += __shfl_xor(sm, d, 32);
        rinv[r] = 1.0f / sm;
    }

    // ---- C-layout -> A-layout conversion via LDS (reuse K region) ----
    __syncthreads();                       // all waves done reading ldsK
    float* attn = ldsK + wave * 4096;      // private 16 x 256 strip per wave
    #pragma unroll
    for (int t = 0; t < 16; ++t) {
        #pragma unroll
        for (int r = 0; r < 8; ++r) {
            attn[(r + half * 8) * 256 + t * 16 + lq] = S[t][r] * rinv[r];
        }
    }

    // ---- O = attn @ V : 64 k-steps x 4 d-tiles of wmma 16x16x4 ----
    v8f o0 = {}, o1 = {}, o2 = {}, o3 = {};
    const float* arow = attn + lq * 256 + half * 2;
    #pragma unroll 8
    for (int jj = 0; jj < 64; ++jj) {
        v2f a2 = *(const v2f*)(arow + jj * 4);
        const float* vb = ldsV + (size_t)(jj * 4 + half * 2) * 64 + lq;
        // batch the 8 LDS reads, then run 4 independent wmma chains
        float b0x = vb[0],  b0y = vb[64];
        float b1x = vb[16], b1y = vb[80];
        float b2x = vb[32], b2y = vb[96];
        float b3x = vb[48], b3y = vb[112];
        v2f b;
        b.x = b0x; b.y = b0y;
        o0 = __builtin_amdgcn_wmma_f32_16x16x4_f32(false, a2, false, b, (short)0, o0, false, false);
        b.x = b1x; b.y = b1y;
        o1 = __builtin_amdgcn_wmma_f32_16x16x4_f32(false, a2, false, b, (short)0, o1, false, false);
        b.x = b2x; b.y = b2y;
        o2 = __builtin_amdgcn_wmma_f32_16x16x4_f32(false, a2, false, b, (short)0, o2, false, false);
        b.x = b3x; b.y = b3y;
        o3 = __builtin_amdgcn_wmma_f32_16x16x4_f32(false, a2, false, b, (short)0, o3, false, false);
    }

    // ---- store output from C-layout: lanes 0-15 write 64B runs ----
    #pragma unroll
    for (int r = 0; r < 8; ++r) {
        float* orow = Og + (size_t)(i0 + r + half * 8) * 64 + lq;
        orow[0]  = o0[r];
        orow[16] = o1[r];
        orow[32] = o2[r];
        orow[48] = o3[r];
    }
}

extern "C" void kernel_launch(void* const* d_in, const int* in_sizes, int n_in,
                              void* d_out, int out_size, void* d_ws, size_t ws_size,
                              hipStream_t stream) {
    const float* q   = (const float*)d_in[0];
    const float* k   = (const float*)d_in[1];
    const float* v   = (const float*)d_in[2];
    const float* pos = (const float*)d_in[3];
    const float* msk = (const float*)d_in[4];
    float* out = (float*)d_out;

    // in_sizes[0] = B*P*H*I*D ; I=256, D=64 fixed by the reference shapes.
    const int nbph = in_sizes[0] / (256 * 64);
    dim3 grid(nbph * 4);   // 4 I-tiles of 64 rows each
    dim3 block(128);       // 4 wave32s
    hipLaunchKernelGGL(wmattn_gfx1250_kernel, grid, block, 0, stream,
                       q, k, v, pos, msk, out);
}